// DGCNN_70342974373894
// MI455X (gfx1250) — compile-verified
//
#include <hip/hip_runtime.h>

#define NPTS 4096
#define BATCH 4
#define KNN 20
#define GFW (16 * KNN)  // 320 edge-feature columns per workgroup
#define EPS 1e-5f
#define NEG_INF (-3.4e38f)

typedef __attribute__((ext_vector_type(2))) float v2f;
typedef __attribute__((ext_vector_type(8))) float v8f;

// D = A(16x4) * B(4x16) + C, fp32 exact.
// (neg_a, A, neg_b, B, c_mod, C, reuse_a, reuse_b)
__device__ __forceinline__ v8f wmma4(v2f a, v2f b, v8f c) {
  return __builtin_amdgcn_wmma_f32_16x16x4_f32(false, a, false, b, (short)0, c,
                                               false, false);
}

// ---------------------------------------------------------------- transpose
// (B,N,3) -> (B,4,N) with channel 3 zeroed (pads kNN K-loop to 4).
__global__ void k_transpose(const float* __restrict__ x, float* __restrict__ xt) {
  int i = blockIdx.x * blockDim.x + threadIdx.x;
  if (i >= BATCH * NPTS) return;
  int b = i / NPTS, n = i % NPTS;
#pragma unroll
  for (int c = 0; c < 3; ++c)
    xt[((size_t)b * 4 + c) * NPTS + n] = x[((size_t)b * NPTS + n) * 3 + c];
  xt[((size_t)b * 4 + 3) * NPTS + n] = 0.f;
}

// ---------------------------------------------------------------- pad W0 64x9 -> 64x12
__global__ void k_padW0(const float* __restrict__ W, float* __restrict__ Wp) {
  int i = blockIdx.x * blockDim.x + threadIdx.x;
  if (i >= 64 * 12) return;
  int co = i / 12, c = i % 12;
  Wp[i] = (c < 9) ? W[co * 9 + c] : 0.f;
}

// ---------------------------------------------------------------- squared norms
__global__ void k_sqnorm(const float* __restrict__ feat, int batchStride,
                         int chanOff, int C, float* __restrict__ xx) {
  int i = blockIdx.x * blockDim.x + threadIdx.x;
  if (i >= BATCH * NPTS) return;
  int b = i / NPTS, n = i % NPTS;
  const float* f = feat + (size_t)b * batchStride + (size_t)chanOff * NPTS + n;
  float s = 0.f;
  for (int c = 0; c < C; ++c) {
    float v = f[(size_t)c * NPTS];
    s += v * v;
  }
  xx[i] = s;
}

// ---------------------------------------------------------------- kNN top-K
// One wave per 16-row block; C must be a multiple of 4 (padded channels = 0).
__global__ void __launch_bounds__(32) k_knn(const float* __restrict__ feat,
                                            int batchStride, int chanOff, int C,
                                            const float* __restrict__ xx,
                                            int* __restrict__ idxOut) {
  __shared__ float dtile[16][17];
  __shared__ float lVal[16][KNN];
  __shared__ int lIdx[16][KNN];
  __shared__ float lMin[16];
  __shared__ int lMinPos[16];

  int wg = blockIdx.x;  // B * N/16
  int b = wg / (NPTS / 16);
  int r0 = (wg % (NPTS / 16)) * 16;
  int lane = threadIdx.x & 31;
  int hf = lane >> 4, l16 = lane & 15;
  const float* f = feat + (size_t)b * batchStride + (size_t)chanOff * NPTS;
  const float* xxb = xx + b * NPTS;

  if (lane < 16) {
    for (int t = 0; t < KNN; ++t) { lVal[lane][t] = NEG_INF; lIdx[lane][t] = 0; }
    lMin[lane] = NEG_INF;
    lMinPos[lane] = 0;
  }
  __syncthreads();

  float xr[8];
#pragma unroll
  for (int e = 0; e < 8; ++e) xr[e] = xxb[r0 + e + 8 * hf];

  for (int jt = 0; jt < NPTS / 16; ++jt) {
    int j0 = jt * 16;
    v8f d = {};
    for (int kc = 0; kc < C; kc += 4) {
      int c0 = kc + 2 * hf;  // lane-half selects K pair (ISA 16x4 A layout)
      const float* fr0 = f + (size_t)c0 * NPTS;
      const float* fr1 = f + (size_t)(c0 + 1) * NPTS;
      v2f a, bv;
      a.x = fr0[r0 + l16];
      a.y = fr1[r0 + l16];
      bv.x = fr0[j0 + l16];
      bv.y = fr1[j0 + l16];
      d = wmma4(a, bv, d);
    }
    float xc = xxb[j0 + l16];
#pragma unroll
    for (int e = 0; e < 8; ++e) {
      int m = e + 8 * hf;  // D layout: vgpr e -> row e / e+8 by lane half
      dtile[m][l16] = 2.f * d[e] - xr[e] - xc;
    }
    __syncthreads();
    if (lane < 16) {
      int r = lane;
      for (int j = 0; j < 16; ++j) {
        float v = dtile[r][j];
        if (v > lMin[r]) {
          int mp = lMinPos[r];
          lVal[r][mp] = v;
          lIdx[r][mp] = j0 + j;
          float mn = lVal[r][0];
          mp = 0;
          for (int t = 1; t < KNN; ++t)
            if (lVal[r][t] < mn) { mn = lVal[r][t]; mp = t; }
          lMin[r] = mn;
          lMinPos[r] = mp;
        }
      }
    }
    __syncthreads();
  }
  if (lane < 16) {
    int* o = idxOut + ((size_t)(b * NPTS + r0 + lane)) * KNN;
    for (int t = 0; t < KNN; ++t) o[t] = lIdx[lane][t];
  }
}

// ---------------------------------------------------------------- EdgeConv
// Workgroup = 16 points, 4 waves.  Columns tiled k-major: tile = one neighbor
// slot across 16 points -> max over K = elementwise register max across the
// 20 accumulator fragments.  W must be pre-padded so cinPad needs no guards.
template <bool FIRST>
__global__ void __launch_bounds__(128) k_edgeconv(
    const float* __restrict__ featIn, int inBatchStride, int inChanOff, int Ch,
    int cinPad, int wStride, const float* __restrict__ W,
    const float* __restrict__ gam, const float* __restrict__ bet,
    const float* __restrict__ mu, const float* __restrict__ var,
    const int* __restrict__ idx, float* __restrict__ catOut, int outChanOff,
    int cout) {
  __shared__ int nbr[16][KNN];
  __shared__ float cfs[128 * 16];  // center features, Ch x 16

  int wg = blockIdx.x;  // B * N/16
  int b = wg / (NPTS / 16);
  int n0 = (wg % (NPTS / 16)) * 16;
  int tid = threadIdx.x;
  int wave = tid >> 5, lane = tid & 31;
  int hf = lane >> 4, l16 = lane & 15;

  const float* fin =
      featIn + (size_t)b * inBatchStride + (size_t)inChanOff * NPTS;

  for (int t = tid; t < 16 * KNN; t += blockDim.x) {
    int p = t / KNN, k = t % KNN;
    nbr[p][k] = idx[((size_t)(b * NPTS + n0 + p)) * KNN + k];
  }
  for (int t = tid; t < Ch * 16; t += blockDim.x) {
    int c = t >> 4, p = t & 15;
    cfs[t] = fin[(size_t)c * NPTS + n0 + p];
  }
  __syncthreads();

  // FIRST: stage full 12 x 320 edge-feature tile (diff | cross | c | 0-pad).
  if constexpr (FIRST) {
    __shared__ float gfs[12 * GFW];
    for (int t = tid; t < GFW; t += blockDim.x) {
      int p = t / KNN;
      int g = nbr[p][t % KNN];
      float c0v = cfs[p], c1v = cfs[16 + p], c2v = cfs[32 + p];
      float n0v = fin[g], n1v = fin[NPTS + g], n2v = fin[2 * NPTS + g];
      gfs[0 * GFW + t] = n0v - c0v;
      gfs[1 * GFW + t] = n1v - c1v;
      gfs[2 * GFW + t] = n2v - c2v;
      gfs[3 * GFW + t] = c1v * n2v - c2v * n1v;  // cross(c, nb)
      gfs[4 * GFW + t] = c2v * n0v - c0v * n2v;
      gfs[5 * GFW + t] = c0v * n1v - c1v * n0v;
      gfs[6 * GFW + t] = c0v;
      gfs[7 * GFW + t] = c1v;
      gfs[8 * GFW + t] = c2v;
      gfs[9 * GFW + t] = 0.f;
      gfs[10 * GFW + t] = 0.f;
      gfs[11 * GFW + t] = 0.f;
    }
    __syncthreads();

    int Mtiles = cout >> 4;
    for (int mt = wave; mt < Mtiles; mt += (int)(blockDim.x >> 5)) {
      float sc[8], sh[8];
#pragma unroll
      for (int e = 0; e < 8; ++e) {
        int co = mt * 16 + e + 8 * hf;
        float s = gam[co] * rsqrtf(var[co] + EPS);
        sc[e] = s;
        sh[e] = bet[co] - mu[co] * s;
      }
      float dmax[8];
#pragma unroll
      for (int e = 0; e < 8; ++e) dmax[e] = NEG_INF;
      const float* Wt = W + (size_t)(mt * 16 + l16) * wStride;

      for (int k = 0; k < KNN; ++k) {
        v8f d = {};
        int jcol = l16 * KNN + k;
        for (int kc = 0; kc < cinPad; kc += 4) {
          int c0 = kc + 2 * hf;
          v2f a = *(const v2f*)(Wt + c0);  // even c0, even stride: 8B aligned
          v2f bv;
          bv.x = gfs[c0 * GFW + jcol];
          bv.y = gfs[(c0 + 1) * GFW + jcol];
          d = wmma4(a, bv, d);
        }
#pragma unroll
        for (int e = 0; e < 8; ++e) dmax[e] = fmaxf(dmax[e], d[e]);
      }

      float* outp = catOut + (size_t)b * 512 * NPTS +
                    (size_t)outChanOff * NPTS + n0 + l16;
#pragma unroll
      for (int e = 0; e < 8; ++e) {
        int co = mt * 16 + e + 8 * hf;
        float y = dmax[e] * sc[e] + sh[e];
        y = (y >= 0.f) ? y : 0.2f * y;
        outp[(size_t)co * NPTS] = y;
      }
    }
  } else {
    int Mtiles = cout >> 4;
    for (int mt = wave; mt < Mtiles; mt += (int)(blockDim.x >> 5)) {
      float sc[8], sh[8];
#pragma unroll
      for (int e = 0; e < 8; ++e) {
        int co = mt * 16 + e + 8 * hf;
        float s = gam[co] * rsqrtf(var[co] + EPS);
        sc[e] = s;
        sh[e] = bet[co] - mu[co] * s;
      }
      float dmax[8];
#pragma unroll
      for (int e = 0; e < 8; ++e) dmax[e] = NEG_INF;
      const float* Wt = W + (size_t)(mt * 16 + l16) * wStride;

      for (int k = 0; k < KNN; ++k) {
        v8f d = {};
        int gidx = nbr[l16][k];
        for (int kc = 0; kc < cinPad; kc += 4) {
          int c0 = kc + 2 * hf;
          v2f a = *(const v2f*)(Wt + c0);
          // Branchless edge feature: channels [0,Ch) = nb - c, [Ch,2Ch) = c.
          // Unconditional gather stays inside the 512-ch cat buffer.
          bool lo0 = c0 < Ch, lo1 = (c0 + 1) < Ch;
          float nb0 = fin[(size_t)c0 * NPTS + gidx];
          float nb1 = fin[(size_t)(c0 + 1) * NPTS + gidx];
          float cf0 = cfs[(lo0 ? c0 : c0 - Ch) * 16 + l16];
          float cf1 = cfs[(lo1 ? c0 + 1 : c0 + 1 - Ch) * 16 + l16];
          v2f bv;
          bv.x = lo0 ? (nb0 - cf0) : cf0;
          bv.y = lo1 ? (nb1 - cf1) : cf1;
          d = wmma4(a, bv, d);
        }
#pragma unroll
        for (int e = 0; e < 8; ++e) dmax[e] = fmaxf(dmax[e], d[e]);
      }

      float* outp = catOut + (size_t)b * 512 * NPTS +
                    (size_t)outChanOff * NPTS + n0 + l16;
#pragma unroll
      for (int e = 0; e < 8; ++e) {
        int co = mt * 16 + e + 8 * hf;
        float y = dmax[e] * sc[e] + sh[e];
        y = (y >= 0.f) ? y : 0.2f * y;
        outp[(size_t)co * NPTS] = y;
      }
    }
  }
}

// ---------------------------------------------------------------- Conv1d + global max
__global__ void __launch_bounds__(128) k_conv1d_pool(
    const float* __restrict__ catIn, const float* __restrict__ W,
    const float* __restrict__ gam, const float* __restrict__ bet,
    const float* __restrict__ mu, const float* __restrict__ var,
    float* __restrict__ pooled) {
  __shared__ float ytile[4][16][17];
  int wg = blockIdx.x;  // B * 32
  int b = wg >> 5;
  int mt = wg & 31;
  int tid = threadIdx.x, wave = tid >> 5, lane = tid & 31;
  int hf = lane >> 4, l16 = lane & 15;
  const float* fin = catIn + (size_t)b * 512 * NPTS;
  const float* Wt = W + (size_t)(mt * 16 + l16) * 512;

  float dmax[8];
#pragma unroll
  for (int e = 0; e < 8; ++e) dmax[e] = NEG_INF;

  for (int nt = wave; nt < NPTS / 16; nt += 4) {
    int n0 = nt * 16;
    v8f d = {};
    for (int kc = 0; kc < 512; kc += 4) {
      int c0 = kc + 2 * hf;
      v2f a = *(const v2f*)(Wt + c0);
      v2f bv;
      bv.x = fin[(size_t)c0 * NPTS + n0 + l16];
      bv.y = fin[(size_t)(c0 + 1) * NPTS + n0 + l16];
      d = wmma4(a, bv, d);
    }
#pragma unroll
    for (int e = 0; e < 8; ++e) dmax[e] = fmaxf(dmax[e], d[e]);
  }
#pragma unroll
  for (int e = 0; e < 8; ++e) ytile[wave][e + 8 * hf][l16] = dmax[e];
  __syncthreads();
  if (tid < 16) {
    float v = NEG_INF;
    for (int n = 0; n < 16; ++n) {
      float v01 = fmaxf(ytile[0][tid][n], ytile[1][tid][n]);
      float v23 = fmaxf(ytile[2][tid][n], ytile[3][tid][n]);
      v = fmaxf(v, fmaxf(v01, v23));
    }
    int co = mt * 16 + tid;
    float s = gam[co] * rsqrtf(var[co] + EPS);
    float y = v * s + (bet[co] - mu[co] * s);
    y = (y >= 0.f) ? y : 0.2f * y;
    pooled[b * 512 + co] = y;
  }
}

// ---------------------------------------------------------------- head
__global__ void __launch_bounds__(128) k_head(const float* __restrict__ pooled,
                                              const float* __restrict__ We,
                                              const float* __restrict__ Wh,
                                              const float* __restrict__ bh,
                                              float* __restrict__ out) {
  __shared__ float emb[128];
  int b = blockIdx.x, t = threadIdx.x;
  float s = 0.f;
  for (int c = 0; c < 512; ++c) s += pooled[b * 512 + c] * We[t * 512 + c];
  emb[t] = s;
  __syncthreads();
  float o = bh[t];
  for (int c = 0; c < 128; ++c) o += emb[c] * Wh[t * 128 + c];
  out[b * 128 + t] = o;
}

// ---------------------------------------------------------------- launch
extern "C" void kernel_launch(void* const* d_in, const int* in_sizes, int n_in,
                              void* d_out, int out_size, void* d_ws,
                              size_t ws_size, hipStream_t stream) {
  (void)in_sizes; (void)n_in; (void)out_size; (void)ws_size;
  const float* x = (const float*)d_in[0];
  const float *W[5], *G[5], *Bt[5], *M[5], *V[5];
  for (int li = 0; li < 5; ++li) {
    W[li] = (const float*)d_in[1 + li * 5 + 0];
    G[li] = (const float*)d_in[1 + li * 5 + 1];
    Bt[li] = (const float*)d_in[1 + li * 5 + 2];
    M[li] = (const float*)d_in[1 + li * 5 + 3];
    V[li] = (const float*)d_in[1 + li * 5 + 4];
  }
  const float* We = (const float*)d_in[26];
  const float* Wh = (const float*)d_in[27];
  const float* bh = (const float*)d_in[28];
  float* out = (float*)d_out;

  char* ws = (char*)d_ws;
  size_t off = 0;
  auto alloc = [&](size_t bytes) {
    void* p = ws + off;
    off = (off + bytes + 255) & ~(size_t)255;
    return p;
  };
  float* xt = (float*)alloc(sizeof(float) * BATCH * 4 * NPTS);   // 4th ch = 0
  float* cat = (float*)alloc(sizeof(float) * BATCH * 512 * NPTS);
  float* xx = (float*)alloc(sizeof(float) * BATCH * NPTS);
  int* idx = (int*)alloc(sizeof(int) * BATCH * NPTS * KNN);
  float* pooled = (float*)alloc(sizeof(float) * BATCH * 512);
  float* w0pad = (float*)alloc(sizeof(float) * 64 * 12);

  int nPts = BATCH * NPTS;
  int wgTile = BATCH * (NPTS / 16);

  k_transpose<<<(nPts + 255) / 256, 256, 0, stream>>>(x, xt);
  k_padW0<<<3, 256, 0, stream>>>(W[0], w0pad);

  // ---- block 0 (coords, C=3 padded to 4, cross term)
  k_sqnorm<<<(nPts + 255) / 256, 256, 0, stream>>>(xt, 4 * NPTS, 0, 3, xx);
  k_knn<<<wgTile, 32, 0, stream>>>(xt, 4 * NPTS, 0, 4, xx, idx);
  k_edgeconv<true><<<wgTile, 128, 0, stream>>>(xt, 4 * NPTS, 0, 3, 12, 12,
                                               w0pad, G[0], Bt[0], M[0], V[0],
                                               idx, cat, 0, 64);
  // ---- block 1 (in ch [0,64) -> out ch [64,128))
  k_sqnorm<<<(nPts + 255) / 256, 256, 0, stream>>>(cat, 512 * NPTS, 0, 64, xx);
  k_knn<<<wgTile, 32, 0, stream>>>(cat, 512 * NPTS, 0, 64, xx, idx);
  k_edgeconv<false><<<wgTile, 128, 0, stream>>>(cat, 512 * NPTS, 0, 64, 128,
                                                128, W[1], G[1], Bt[1], M[1],
                                                V[1], idx, cat, 64, 64);
  // ---- block 2 (in ch [64,128) -> out ch [128,256))
  k_sqnorm<<<(nPts + 255) / 256, 256, 0, stream>>>(cat, 512 * NPTS, 64, 64, xx);
  k_knn<<<wgTile, 32, 0, stream>>>(cat, 512 * NPTS, 64, 64, xx, idx);
  k_edgeconv<false><<<wgTile, 128, 0, stream>>>(cat, 512 * NPTS, 64, 64, 128,
                                                128, W[2], G[2], Bt[2], M[2],
                                                V[2], idx, cat, 128, 128);
  // ---- block 3 (in ch [128,256) -> out ch [256,512))
  k_sqnorm<<<(nPts + 255) / 256, 256, 0, stream>>>(cat, 512 * NPTS, 128, 128, xx);
  k_knn<<<wgTile, 32, 0, stream>>>(cat, 512 * NPTS, 128, 128, xx, idx);
  k_edgeconv<false><<<wgTile, 128, 0, stream>>>(cat, 512 * NPTS, 128, 128, 256,
                                                256, W[3], G[3], Bt[3], M[3],
                                                V[3], idx, cat, 256, 256);
  // ---- block 4: 512->512 conv + global max pool
  k_conv1d_pool<<<BATCH * 32, 128, 0, stream>>>(cat, W[4], G[4], Bt[4], M[4],
                                                V[4], pooled);
  // ---- embedding head
  k_head<<<BATCH, 128, 0, stream>>>(pooled, We, Wh, bh, out);
}